// DomainTransformFilter_9689446219846
// MI455X (gfx1250) — compile-verified
//
#include <hip/hip_runtime.h>

// ---------------------------------------------------------------------------
// Domain-transform filter for MI455X (gfx1250, wave32).
// 61-tap separable Gaussian conv expressed as banded-Toeplitz matmul on
// V_WMMA_F32_16X16X4_F32: D[16x16] = sum_{c=0..18} A[16x4] * B[4x16].
// ---------------------------------------------------------------------------

typedef float v2f __attribute__((ext_vector_type(2)));
typedef float v8f __attribute__((ext_vector_type(8)));

#define Bsz 8
#define Csz 3
#define Hsz 1024
#define Wsz 1024
#define HW (Hsz * Wsz)
#define NPLANE (Bsz * Csz)
#define KS 61
#define HK 30
#define PATCH 76          // 16 outputs + 60 halo
#define NCHUNK 19         // 76 / 4
#define SIGMA_R 0.5f
#define EPS 1e-8f

// ---- per-wave Gaussian helpers (uniform across lanes; ~trivial setup cost) --
__device__ __forceinline__ float gsum_inv(float inv2s2) {
    float s = 0.0f;
    for (int t = 0; t < KS; ++t) {
        float d = (float)(t - HK);
        s += __expf(-d * d * inv2s2);
    }
    return 1.0f / s;
}

__device__ __forceinline__ float gval(int t, float inv2s2, float inv_sum) {
    float d = (float)(t - HK);
    float v = __expf(-d * d * inv2s2) * inv_sum;
    return (t >= 0 && t < KS) ? v : 0.0f;
}

// ---------------------------------------------------------------------------
// Kernel 1: edge weights.  sx = wx/(wx+eps) (from W-gradient of guide),
//           sy = wy/(wy+eps) (from H-gradient).  guide = channel mean.
// ---------------------------------------------------------------------------
__global__ void __launch_bounds__(256)
k_weights(const float* __restrict__ in, float* __restrict__ sx,
          float* __restrict__ sy) {
    int idx = blockIdx.x * 256 + threadIdx.x;
    if (idx >= Bsz * HW) return;
    int b   = idx / HW;
    int rem = idx - b * HW;
    int y   = rem / Wsz;
    int x   = rem - y * Wsz;

    const float* p = in + (size_t)b * Csz * HW;
    auto guide = [&](int yy, int xx) -> float {
        int o = yy * Wsz + xx;
        return (p[o] + p[o + HW] + p[o + 2 * HW]) * (1.0f / 3.0f);
    };
    // jnp.pad(grad, 'edge'): last pixel replicates the last *gradient*
    int x1 = (x + 1 < Wsz) ? x + 1 : Wsz - 1;
    int x0 = (x == Wsz - 1) ? Wsz - 2 : x;
    int y1 = (y + 1 < Hsz) ? y + 1 : Hsz - 1;
    int y0 = (y == Hsz - 1) ? Hsz - 2 : y;

    float gx = guide(y, x1) - guide(y, x0);
    float gy = guide(y1, x) - guide(y0, x);
    const float inv2r2 = 1.0f / (2.0f * SIGMA_R * SIGMA_R);
    float wx = __expf(-gx * gx * inv2r2);
    float wy = __expf(-gy * gy * inv2r2);
    sx[idx] = wx / (wx + EPS);
    sy[idx] = wy / (wy + EPS);
}

// ---------------------------------------------------------------------------
// Kernel 2: horizontal conv (axis=W) * sy.
// A = image data (lanes = rows) fed from an LDS-staged strip via ds_load_b64,
// B = Toeplitz g[j-x] in registers.  D lanes = columns -> coalesced stores.
// ---------------------------------------------------------------------------
#define HX_STRIP 128               // output columns per wave
#define HX_LCOLS (HX_STRIP + 60)   // 188 staged columns
#define HX_WAVES 4

__global__ void __launch_bounds__(32 * HX_WAVES)
k_hconv(const float* __restrict__ in, const float* __restrict__ sy,
        float* __restrict__ out, float inv2s2) {
    __shared__ float lds[HX_WAVES][16 * HX_LCOLS];   // 48128 B

    int lane  = threadIdx.x & 31;
    int wave  = threadIdx.x >> 5;
    int lo    = lane & 15;          // A: row m    / B: col n    / D: col n
    int hi    = lane >> 4;          // K half (k = 2*hi + vgpr)
    int plane = blockIdx.z;
    int b     = plane / Csz;
    int rbase = blockIdx.y * 16;
    int c0    = (blockIdx.x * HX_WAVES + wave) * HX_STRIP;

    // ---- stage 16 x 188 strip (replicate-clamped cols), coalesced ----------
    const float* src = in + (size_t)plane * HW;
    float* lw = lds[wave];
    for (int r = 0; r < 16; ++r) {
        const float* rowp = src + (size_t)(rbase + r) * Wsz;
        for (int j = lane; j < HX_LCOLS; j += 32) {
            int gc = c0 + j - HK;
            gc = gc < 0 ? 0 : (gc > Wsz - 1 ? Wsz - 1 : gc);
            lw[r * HX_LCOLS + j] = rowp[gc];
        }
    }
    __syncthreads();

    // ---- constant Toeplitz operand: B_c[k,n] = g[4c + k - n] ---------------
    float inv_sum = gsum_inv(inv2s2);
    v2f tb[NCHUNK];
#pragma unroll
    for (int c = 0; c < NCHUNK; ++c) {
        int base = 4 * c + 2 * hi - lo;
        tb[c].x = gval(base,     inv2s2, inv_sum);
        tb[c].y = gval(base + 1, inv2s2, inv_sum);
    }

    const float* syp = sy + (size_t)b * HW;
    float* dst = out + (size_t)plane * HW;

    for (int t = 0; t < HX_STRIP / 16; ++t) {
        int toff = t * 16;
        v8f acc = {};
#pragma unroll
        for (int c = 0; c < NCHUNK; ++c) {
            // A_c[m,k] = patch[4c+k, m]; element index even -> 8B aligned b64
            const v2f a = *(const v2f*)&lw[lo * HX_LCOLS + toff + 4 * c + 2 * hi];
            acc = __builtin_amdgcn_wmma_f32_16x16x4_f32(
                false, a, false, tb[c], (short)0, acc, false, false);
        }
        int col = c0 + toff + lo;
#pragma unroll
        for (int v = 0; v < 8; ++v) {
            int row = rbase + v + 8 * hi;
            size_t o = (size_t)row * Wsz + col;
            dst[o] = acc[v] * syp[o];
        }
    }
}

// ---------------------------------------------------------------------------
// Kernel 3: vertical conv (axis=H) * sx.
// B = image data (lanes = columns -> coalesced 64B loads), A = Toeplitz.
// ---------------------------------------------------------------------------
#define VX_TILES 16   // 16 tiles x 16 cols = 256 output columns per wave

__global__ void __launch_bounds__(128)
k_vconv(const float* __restrict__ in, const float* __restrict__ sx,
        float* __restrict__ out, float inv2s2) {
    int lane  = threadIdx.x & 31;
    int wave  = threadIdx.x >> 5;          // column group 0..3
    int lo    = lane & 15;
    int hi    = lane >> 4;
    int plane = blockIdx.z;
    int b     = plane / Csz;
    int rbase = blockIdx.y * 16;

    const float* src = in + (size_t)plane * HW;
    const float* sxp = sx + (size_t)b * HW;
    float* dst = out + (size_t)plane * HW;

    // ---- constant Toeplitz operand: A_c[m,k] = g[4c + k - m] ---------------
    float inv_sum = gsum_inv(inv2s2);
    v2f ta[NCHUNK];
#pragma unroll
    for (int c = 0; c < NCHUNK; ++c) {
        int base = 4 * c + 2 * hi - lo;
        ta[c].x = gval(base,     inv2s2, inv_sum);
        ta[c].y = gval(base + 1, inv2s2, inv_sum);
    }

    for (int t = 0; t < VX_TILES; ++t) {
        int col = wave * (VX_TILES * 16) + t * 16 + lo;
        v8f acc = {};
#pragma unroll
        for (int c = 0; c < NCHUNK; ++c) {
            int j  = 4 * c + 2 * hi;
            int r0 = rbase + j - HK;
            int ra = r0 < 0 ? 0 : (r0 > Hsz - 1 ? Hsz - 1 : r0);
            int r1 = r0 + 1 < 0 ? 0 : (r0 + 1 > Hsz - 1 ? Hsz - 1 : r0 + 1);
            v2f bb;
            bb.x = src[(size_t)ra * Wsz + col];   // lanes 0-15 / 16-31: 64B runs
            bb.y = src[(size_t)r1 * Wsz + col];
            acc = __builtin_amdgcn_wmma_f32_16x16x4_f32(
                false, ta[c], false, bb, (short)0, acc, false, false);
        }
#pragma unroll
        for (int v = 0; v < 8; ++v) {
            int row = rbase + v + 8 * hi;
            size_t o = (size_t)row * Wsz + col;
            dst[o] = acc[v] * sxp[o];
        }
    }
}

// ---------------------------------------------------------------------------
// Driver: weights, then 3 x (H-conv -> tmp, V-conv -> d_out).
// ws layout: sx [8*HW] | sy [8*HW] | tmp image [24*HW]   (~168 MB)
// ---------------------------------------------------------------------------
extern "C" void kernel_launch(void* const* d_in, const int* in_sizes, int n_in,
                              void* d_out, int out_size, void* d_ws, size_t ws_size,
                              hipStream_t stream) {
    (void)in_sizes; (void)n_in; (void)out_size; (void)ws_size;
    const float* in  = (const float*)d_in[0];
    float*       out = (float*)d_out;

    float* sx  = (float*)d_ws;
    float* sy  = sx + (size_t)Bsz * HW;
    float* tmp = sy + (size_t)Bsz * HW;

    k_weights<<<dim3((Bsz * HW + 255) / 256), 256, 0, stream>>>(in, sx, sy);

    const float* cur = in;
    for (int i = 0; i < 3; ++i) {
        float sigma  = 10.0f * (float)(1 << i);
        float inv2s2 = 1.0f / (2.0f * sigma * sigma);
        k_hconv<<<dim3(Wsz / (HX_STRIP * HX_WAVES), Hsz / 16, NPLANE),
                  32 * HX_WAVES, 0, stream>>>(cur, sy, tmp, inv2s2);
        k_vconv<<<dim3(1, Hsz / 16, NPLANE), 128, 0, stream>>>(tmp, sx, out, inv2s2);
        cur = out;
    }
}